// PointPillarsBEV_88948772700341
// MI455X (gfx1250) — compile-verified
//
#include <hip/hip_runtime.h>

// PointPillars BEV: PFN (32x10 @ 10x64 per pillar, f32 WMMA) + ReLU + max-pool
// over points + atomic scatter-add into (B, C, 512, 512) canvas.
//
// Memory-bound: ~320 MB traffic @ 23.3 TB/s. Kernel 1 zero-fills the 256 MB
// canvas with non-temporal b128 stores; kernel 2 fuses PFN + reduce + scatter
// (one wave32 per pillar, 24x V_WMMA_F32_16X16X4_F32 per pillar, exact f32).

typedef __attribute__((ext_vector_type(2))) float v2f;
typedef __attribute__((ext_vector_type(4))) float v4f;
typedef __attribute__((ext_vector_type(8))) float v8f;

#define BEV_H 512
#define BEV_W 512
#define NB    4
#define NP    12000
#define NPTS  32
#define ND    10
#define NC    64
#define KPAD  12   // K padded 10 -> 12 (three K=4 chunks)

__global__ void __launch_bounds__(256)
bev_zero_kernel(v4f* __restrict__ out) {
    size_t i = (size_t)blockIdx.x * blockDim.x + threadIdx.x;
    v4f z = {0.f, 0.f, 0.f, 0.f};
    __builtin_nontemporal_store(z, &out[i]);
}

__global__ void __launch_bounds__(128)
pfn_wmma_scatter_kernel(const float* __restrict__ pillars,
                        const int*   __restrict__ coords,
                        const float* __restrict__ W,     // (10, 64) row-major
                        const float* __restrict__ bias,  // (64,)
                        float*       __restrict__ out)   // (B, 64, 512, 512)
{
    __shared__ __align__(16) float smem[4][NPTS][KPAD];

    const int lane = threadIdx.x & 31;
    // wave id must be scalar so the whole kernel is wave-uniform (EXEC all-1s
    // at the WMMAs, as the ISA requires).
    const int wid  = __builtin_amdgcn_readfirstlane((int)(threadIdx.x >> 5));
    const int pillar = blockIdx.x * 4 + wid;   // grid is exactly NB*NP/4 blocks

    const int nloc = lane & 15;   // N (or M) position within a 16-wide tile
    const int hi   = lane >> 4;   // lane-half selector

    // ---- Preload B fragments: Breg[kchunk][ntile], 4x16 f32 layout:
    // VGPR0 holds row K = k0+2*hi, VGPR1 holds row K+1, lanes%16 = N.
    v2f Breg[3][4];
#pragma unroll
    for (int kc = 0; kc < 3; ++kc) {
        const int k = kc * 4 + hi * 2;
#pragma unroll
        for (int t = 0; t < 4; ++t) {
            float bx = (k     < ND) ? W[(k    ) * NC + t * 16 + nloc] : 0.f;
            float by = (k + 1 < ND) ? W[(k + 1) * NC + t * 16 + nloc] : 0.f;
            v2f b; b.x = bx; b.y = by;
            Breg[kc][t] = b;
        }
    }

    // ---- Stage A (32 points x 10 dims, padded to 12) in LDS; lane = point row.
    {
        const float* src = pillars + ((size_t)pillar * NPTS + lane) * ND;
        float r[ND];
#pragma unroll
        for (int k = 0; k < ND; ++k) r[k] = src[k];
#pragma unroll
        for (int k = 0; k < ND; ++k) smem[wid][lane][k] = r[k];
        smem[wid][lane][10] = 0.f;
        smem[wid][lane][11] = 0.f;
    }
    __syncthreads();  // order cross-lane LDS store -> load (uniform barrier)

    // ---- 24x WMMA f32 16x16x4: acc[mtile][ntile] (+= A-tile @ B-tile)
    v8f acc[2][4];
#pragma unroll
    for (int mt = 0; mt < 2; ++mt)
#pragma unroll
        for (int t = 0; t < 4; ++t)
#pragma unroll
            for (int r = 0; r < 8; ++r) acc[mt][t][r] = 0.f;

#pragma unroll
    for (int kc = 0; kc < 3; ++kc) {
        // A 16x4 f32 layout: lane L holds M = mt*16 + L%16,
        // VGPR0 = K = k0 + 2*(L/16), VGPR1 = K+1  -> one ds_load_b64 per tile.
        v2f a[2];
        const int k = kc * 4 + hi * 2;
#pragma unroll
        for (int mt = 0; mt < 2; ++mt) {
            const int m = mt * 16 + nloc;
            a[mt] = *(const v2f*)&smem[wid][m][k];
        }
#pragma unroll
        for (int mt = 0; mt < 2; ++mt)
#pragma unroll
            for (int t = 0; t < 4; ++t)
                acc[mt][t] = __builtin_amdgcn_wmma_f32_16x16x4_f32(
                    /*neg_a=*/false, a[mt],
                    /*neg_b=*/false, Breg[kc][t],
                    /*c_mod=*/(short)0, acc[mt][t],
                    /*reuse_a=*/false, /*reuse_b=*/false);
    }

    // ---- max over the 32 points, + bias, ReLU, atomic scatter.
    // C/D layout: VGPR r -> M = r + 8*hi, lanes%16 -> N. So: max over the 8
    // accumulator regs, shfl_xor(16) to merge lane halves, max across M-tiles.
    const int b = pillar / NP;
    const int y = coords[pillar * 2 + 0];
    const int x = coords[pillar * 2 + 1];
    float* outb = out + (size_t)b * NC * (BEV_H * BEV_W)
                      + (size_t)y * BEV_W + x;

#pragma unroll
    for (int t = 0; t < 4; ++t) {
        float m0 = acc[0][t][0];
        float m1 = acc[1][t][0];
#pragma unroll
        for (int r = 1; r < 8; ++r) {
            m0 = fmaxf(m0, acc[0][t][r]);
            m1 = fmaxf(m1, acc[1][t][r]);
        }
        float v = fmaxf(m0, m1);
        v = fmaxf(v, __shfl_xor(v, 16, 32));          // merge M halves
        v = fmaxf(v + bias[t * 16 + nloc], 0.f);      // bias + ReLU (commute w/ max)
        if (lane < 16) {
            // channel c = t*16 + lane, canvas layout (B, C, H, W)
            atomicAdd(outb + (size_t)(t * 16 + lane) * (BEV_H * BEV_W), v);
        }
    }
}

extern "C" void kernel_launch(void* const* d_in, const int* in_sizes, int n_in,
                              void* d_out, int out_size, void* d_ws, size_t ws_size,
                              hipStream_t stream) {
    const float* pillars = (const float*)d_in[0];   // (4, 12000, 32, 10) f32
    const int*   coords  = (const int*)  d_in[1];   // (4, 12000, 2) i32
    const float* W       = (const float*)d_in[2];   // (10, 64) f32
    const float* bias    = (const float*)d_in[3];   // (64,) f32
    float*       out     = (float*)d_out;           // (4, 64, 512, 512) f32

    // 4*64*512*512 = 67,108,864 floats = 16,777,216 b128 stores
    bev_zero_kernel<<<65536, 256, 0, stream>>>((v4f*)out);

    // one wave32 per pillar, 4 waves per block: 48000 / 4 = 12000 blocks
    pfn_wmma_scatter_kernel<<<(NB * NP) / 4, 128, 0, stream>>>(
        pillars, coords, W, bias, out);
}